// EmbeddingWithDropout_52321291599899
// MI455X (gfx1250) — compile-verified
//
#include <hip/hip_runtime.h>

// Embedding gather + per-row dropout mask.
//   x      : [64, 2048] int32 token ids           (in_sizes[0] = 131072)
//   weight : [100000, 128] fp32 embedding table   (51.2 MB, L2-resident on MI455X)
//   u      : [100000] fp32 per-row uniforms
//   out    : [64, 2048, 128] fp32 = weight[x] * ((u[x] >= 0.1) / 0.9)
//
// Pure bandwidth problem (~102 MB @ 23.3 TB/s ~= 4.4 us). No matmul -> no WMMA.
// Strategy per wave32:
//   - one global_load_b32 fetches all 8 token ids (lane & 7 addressing),
//   - one global_load_b32 gathers the 8 u[] values,
//   - v_readlane extracts per-token row id / scale into SGPRs,
//   - 8 independent global_load_b128 row gathers (saddr base + lane*16),
//   - 8 non-temporal global_store_b128 so the streamed 64 MB output does not
//     evict the weight table from the 192 MB L2.

typedef float v4f __attribute__((ext_vector_type(4)));

#define EMB_DROPOUT 0.1f
#define KEEP_SCALE  (1.0f / (1.0f - EMB_DROPOUT))
#define TOK_PER_WAVE 8

__global__ __launch_bounds__(256) void embed_dropout_gather_kernel(
    const int* __restrict__ x,
    const float* __restrict__ weight,
    const float* __restrict__ u,
    float* __restrict__ out,
    int n_tokens)
{
    const int lane = threadIdx.x & 31;                          // wave32
    // Force the wave id scalar so t0 lives in an SGPR: all bounds checks
    // become scalar branches and all base addresses use saddr forms.
    const int waveInBlk = __builtin_amdgcn_readfirstlane(threadIdx.x >> 5);
    const int wave = blockIdx.x * (blockDim.x >> 5) + waveInBlk;
    const int t0   = wave * TOK_PER_WAVE;
    if (t0 >= n_tokens) return;

    if (t0 + TOK_PER_WAVE <= n_tokens) {
        // ---- fast path: full tile of 8 tokens ----
        // One coalesced load of the 8 consecutive ids (each duplicated 4x).
        const int vidx = x[t0 + (lane & (TOK_PER_WAVE - 1))];
        // One gather of the 8 dropout uniforms (L2 hits, duplicated 4x).
        const float ul  = u[vidx];
        const float scl = (ul >= EMB_DROPOUT) ? KEEP_SCALE : 0.0f;
        const int   sclbits = __builtin_bit_cast(int, scl);

#pragma unroll
        for (int i = 0; i < TOK_PER_WAVE; ++i) {
            // Immediate-lane readlane -> SGPR, no serializing wave ops.
            const int   row = __builtin_amdgcn_readlane(vidx, i);
            const float sc  = __builtin_bit_cast(
                float, __builtin_amdgcn_readlane(sclbits, i));

            // 512 B row = 32 lanes x b128; scalar row base, vector lane offset.
            v4f v = *((const v4f*)(weight + (size_t)row * 128) + lane);
            v *= sc;
            // Write-once output: non-temporal to preserve L2 for the table.
            __builtin_nontemporal_store(
                v, (v4f*)(out + (size_t)(t0 + i) * 128) + lane);
        }
    } else {
        // ---- tail (never taken for 131072 tokens, kept for generality) ----
        for (int t = t0; t < n_tokens; ++t) {
            const int   row = __builtin_amdgcn_readfirstlane(x[t]);
            const float sc  = (u[row] >= EMB_DROPOUT) ? KEEP_SCALE : 0.0f;
            v4f v = *((const v4f*)(weight + (size_t)row * 128) + lane);
            v *= sc;
            __builtin_nontemporal_store(
                v, (v4f*)(out + (size_t)t * 128) + lane);
        }
    }
}

extern "C" void kernel_launch(void* const* d_in, const int* in_sizes, int n_in,
                              void* d_out, int out_size, void* d_ws, size_t ws_size,
                              hipStream_t stream)
{
    const int*   x      = (const int*)d_in[0];
    const float* weight = (const float*)d_in[1];
    const float* u      = (const float*)d_in[2];
    float*       out    = (float*)d_out;

    const int n_tokens = in_sizes[0];                        // 64 * 2048 = 131072

    const int threads          = 256;                        // 8 wave32 waves
    const int tokens_per_block = (threads / 32) * TOK_PER_WAVE;  // 64
    const int blocks = (n_tokens + tokens_per_block - 1) / tokens_per_block;

    embed_dropout_gather_kernel<<<blocks, threads, 0, stream>>>(
        x, weight, u, out, n_tokens);
}